// SuperQ_20151986552980
// MI455X (gfx1250) — compile-verified
//
#include <hip/hip_runtime.h>
#include <math.h>
#include <stdint.h>

#define N_SQ_MAX  64
#define PARAM_F   20            // 5 x float4 per primitive (80 B, 16B-aligned)
#define SQ_EPS    1e-6f
#define SQ_EPS2   (1e-6f * 1e-6f)
#define SQ_TRUNC  0.1f
#define MIN_E     0.1f
#define MAX_E     1.9f

// ---------------------------------------------------------------------------
// Kernel 1: fold all point-independent transcendentals into a param table.
// params[i] (20 floats): r00 r01 r02 r10 | r11 r12 r20 r21 | r22 t0 t1 t2 |
//                        1/sx^2 1/sy^2 1/sz^2 1/e1 | 1/e2 e2/e1 -e1/2 pad
// ---------------------------------------------------------------------------
__global__ void sq_prep_kernel(const float* __restrict__ raw_scale,
                               const float* __restrict__ raw_exps,
                               const float* __restrict__ raw_rot,
                               const float* __restrict__ trans,
                               float* __restrict__ params, int nsq) {
  int i = (int)(blockIdx.x * blockDim.x + threadIdx.x);
  if (i >= nsq) return;
  float isx2 = expf(-2.f * raw_scale[3 * i + 0]);   // (x/s)^2 = x^2 * e^{-2 raw}
  float isy2 = expf(-2.f * raw_scale[3 * i + 1]);
  float isz2 = expf(-2.f * raw_scale[3 * i + 2]);
  float e1 = MIN_E + (MAX_E - MIN_E) / (1.f + expf(-raw_exps[2 * i + 0]));
  float e2 = MIN_E + (MAX_E - MIN_E) / (1.f + expf(-raw_exps[2 * i + 1]));
  float w = raw_rot[4 * i + 0], x = raw_rot[4 * i + 1];
  float y = raw_rot[4 * i + 2], z = raw_rot[4 * i + 3];
  float inv = rsqrtf(w * w + x * x + y * y + z * z);
  w *= inv; x *= inv; y *= inv; z *= inv;
  float r00 = 1.f - 2.f * (y * y + z * z), r01 = 2.f * (x * y - w * z), r02 = 2.f * (x * z + w * y);
  float r10 = 2.f * (x * y + w * z), r11 = 1.f - 2.f * (x * x + z * z), r12 = 2.f * (y * z - w * x);
  float r20 = 2.f * (x * z - w * y), r21 = 2.f * (y * z + w * x), r22 = 1.f - 2.f * (x * x + y * y);
  float* P = params + i * PARAM_F;
  P[0]  = r00;  P[1]  = r01;  P[2]  = r02;  P[3]  = r10;
  P[4]  = r11;  P[5]  = r12;  P[6]  = r20;  P[7]  = r21;
  P[8]  = r22;  P[9]  = trans[3 * i + 0]; P[10] = trans[3 * i + 1]; P[11] = trans[3 * i + 2];
  P[12] = isx2; P[13] = isy2; P[14] = isz2; P[15] = 1.f / e1;
  P[16] = 1.f / e2; P[17] = e2 / e1; P[18] = -0.5f * e1; P[19] = 0.f;
}

// ---------------------------------------------------------------------------
// Value-only SDF: sign-clamp folded into max(Xc^2, eps^2) since x enters only
// as x^2 here. 10 exp2/log2 + 1 sqrt + ~30 VALU per pair.
// ---------------------------------------------------------------------------
__device__ __forceinline__ float sq_val(const float4& f0, const float4& f1,
                                        const float4& f2, const float4& f3,
                                        const float4& f4,
                                        float px, float py, float pz) {
  float dx = px - f2.y, dy = py - f2.z, dz = pz - f2.w;
  float Xcx = f0.x * dx + f0.w * dy + f1.z * dz;   // columns of R
  float Xcy = f0.y * dx + f1.x * dy + f1.w * dz;
  float Xcz = f0.z * dx + f1.y * dy + f2.x * dz;
  float x2 = fmaxf(Xcx * Xcx, SQ_EPS2);            // == max(|Xc|,eps)^2
  float y2 = fmaxf(Xcy * Xcy, SQ_EPS2);
  float z2 = fmaxf(Xcz * Xcz, SQ_EPS2);
  float t1 = exp2f(f4.x * log2f(x2 * f3.x));       // (x^2/sx^2)^(1/e2)
  float t2 = exp2f(f4.x * log2f(y2 * f3.y));
  float t3 = exp2f(f3.w * log2f(z2 * f3.z));       // (z^2/sz^2)^(1/e1)
  float A  = t1 + t2 + SQ_EPS;
  float Ae = exp2f(f4.y * log2f(A));               // A^(e2/e1)
  float B  = Ae + t3;
  float f  = exp2f(f4.z * log2f(B));               // B^(-e1/2)
  float r0 = sqrtf(x2 + y2 + z2);
  float s  = r0 * (1.f - f);
  return fminf(fmaxf(s, -SQ_TRUNC), SQ_TRUNC);     // truncate before min-reduce
}

// ---------------------------------------------------------------------------
// Analytic point-gradient of the (unclipped) SDF — run once per point for the
// winning primitive only.  ds/dX = (X/r0)(1-f) + (r0 f/B)[P t1/x, P t2/y, t3/z]
// with P = A^(e2/e1-1) = Ae/A, masked by |Xc|>eps, then grad_p = R * grad_X.
// ---------------------------------------------------------------------------
__device__ __forceinline__ void sq_grad(const float4* __restrict__ q,
                                        float px, float py, float pz,
                                        float g[3]) {
  float4 f0 = q[0], f1 = q[1], f2 = q[2], f3 = q[3], f4 = q[4];
  float dx = px - f2.y, dy = py - f2.z, dz = pz - f2.w;
  float Xcx = f0.x * dx + f0.w * dy + f1.z * dz;
  float Xcy = f0.y * dx + f1.x * dy + f1.w * dz;
  float Xcz = f0.z * dx + f1.y * dy + f2.x * dz;
  float x = (Xcx > 0.f ? 1.f : -1.f) * fmaxf(fabsf(Xcx), SQ_EPS);
  float y = (Xcy > 0.f ? 1.f : -1.f) * fmaxf(fabsf(Xcy), SQ_EPS);
  float z = (Xcz > 0.f ? 1.f : -1.f) * fmaxf(fabsf(Xcz), SQ_EPS);
  float x2 = x * x, y2 = y * y, z2 = z * z;
  float t1 = exp2f(f4.x * log2f(x2 * f3.x));
  float t2 = exp2f(f4.x * log2f(y2 * f3.y));
  float t3 = exp2f(f3.w * log2f(z2 * f3.z));
  float A  = t1 + t2 + SQ_EPS;
  float Ae = exp2f(f4.y * log2f(A));
  float B  = Ae + t3;
  float f  = exp2f(f4.z * log2f(B));
  float r0 = sqrtf(x2 + y2 + z2);
  float one_mf = 1.f - f;
  float cm = r0 * f / B;
  float Pf = Ae / A;
  float ir = 1.f / r0;
  float gx = x * one_mf * ir + cm * Pf * t1 / x;
  float gy = y * one_mf * ir + cm * Pf * t2 / y;
  float gz = z * one_mf * ir + cm * t3 / z;
  gx = (fabsf(Xcx) > SQ_EPS) ? gx : 0.f;           // d max(|Xc|,eps)/dXc
  gy = (fabsf(Xcy) > SQ_EPS) ? gy : 0.f;
  gz = (fabsf(Xcz) > SQ_EPS) ? gz : 0.f;
  g[0] = f0.x * gx + f0.y * gy + f0.z * gz;        // rows of R
  g[1] = f0.w * gx + f1.x * gy + f1.y * gz;
  g[2] = f1.z * gx + f1.w * gy + f2.x * gz;
}

// ---------------------------------------------------------------------------
// Kernel 2: two points per lane (doubles independent trans chains, halves DS
// traffic per point); params staged to LDS with CDNA5 async global->LDS copy.
// ---------------------------------------------------------------------------
__launch_bounds__(256)
__global__ void sq_main_kernel(const float* __restrict__ params_g,
                               const float* __restrict__ pts,
                               float* __restrict__ out,
                               int npts, int nsq) {
  __shared__ float sp[N_SQ_MAX * PARAM_F];  // 5120 B
  const int nflt = nsq * PARAM_F;
  {
    // CDNA5 async staging path (ASYNCcnt): no VGPR round-trip for the table
    uint32_t lds_base = (uint32_t)(uintptr_t)(&sp[0]);
    unsigned long long gb = (unsigned long long)(uintptr_t)params_g;
    for (int k = (int)threadIdx.x; k < nflt; k += (int)blockDim.x) {
      uint32_t off = (uint32_t)k * 4u;
      asm volatile("global_load_async_to_lds_b32 %0, %1, %2"
                   :
                   : "v"(lds_base + off), "v"(off), "s"(gb)
                   : "memory");
    }
    asm volatile("s_wait_asynccnt 0" ::: "memory");
  }
  __syncthreads();

  const int tid  = (int)threadIdx.x;
  const int base = (int)blockIdx.x * 512;
  const int pidA = base + tid;          // coalesced pair of point slabs
  const int pidB = base + 256 + tid;
  const bool hasA = pidA < npts;
  const bool hasB = pidB < npts;

  float pAx = 0.f, pAy = 0.f, pAz = 0.f, pBx = 0.f, pBy = 0.f, pBz = 0.f;
  if (hasA) { pAx = pts[3 * pidA + 0]; pAy = pts[3 * pidA + 1]; pAz = pts[3 * pidA + 2]; }
  if (hasB) { pBx = pts[3 * pidB + 0]; pBy = pts[3 * pidB + 1]; pBz = pts[3 * pidB + 2]; }

  float bestA = 3.402823466e38f, bestB = 3.402823466e38f;
  int   biA = 0, biB = 0;
  #pragma unroll 2
  for (int i = 0; i < nsq; ++i) {
    const float4* q = (const float4*)(sp + i * PARAM_F);
    float4 f0 = q[0], f1 = q[1], f2 = q[2], f3 = q[3], f4 = q[4];
    float sA = sq_val(f0, f1, f2, f3, f4, pAx, pAy, pAz);
    float sB = sq_val(f0, f1, f2, f3, f4, pBx, pBy, pBz);
    if (sA < bestA) { bestA = sA; biA = i; }   // strict <  ==  argmin first-min
    if (sB < bestB) { bestB = sB; biB = i; }
  }

  float* no = out + npts;
  if (hasA) {
    float g[3];
    sq_grad((const float4*)(sp + biA * PARAM_F), pAx, pAy, pAz, g);
    float inn = 1.f / fmaxf(sqrtf(g[0] * g[0] + g[1] * g[1] + g[2] * g[2]), 1e-12f);
    out[pidA] = bestA;
    no[3 * pidA + 0] = g[0] * inn;
    no[3 * pidA + 1] = g[1] * inn;
    no[3 * pidA + 2] = g[2] * inn;
  }
  if (hasB) {
    float g[3];
    sq_grad((const float4*)(sp + biB * PARAM_F), pBx, pBy, pBz, g);
    float inn = 1.f / fmaxf(sqrtf(g[0] * g[0] + g[1] * g[1] + g[2] * g[2]), 1e-12f);
    out[pidB] = bestB;
    no[3 * pidB + 0] = g[0] * inn;
    no[3 * pidB + 1] = g[1] * inn;
    no[3 * pidB + 2] = g[2] * inn;
  }
}

// ---------------------------------------------------------------------------
extern "C" void kernel_launch(void* const* d_in, const int* in_sizes, int n_in,
                              void* d_out, int out_size, void* d_ws, size_t ws_size,
                              hipStream_t stream) {
  const float* raw_scale = (const float*)d_in[0];
  const float* raw_exps  = (const float*)d_in[1];
  const float* raw_rot   = (const float*)d_in[2];
  const float* trans     = (const float*)d_in[3];
  const float* pts       = (const float*)d_in[4];
  int nsq  = in_sizes[0] / 3;   // 64
  int npts = in_sizes[4] / 3;   // 262144
  float* params = (float*)d_ws; // nsq * 20 floats = 5120 B

  sq_prep_kernel<<<(nsq + 63) / 64, 64, 0, stream>>>(raw_scale, raw_exps, raw_rot,
                                                     trans, params, nsq);
  const int block = 256;
  const int grid  = (npts + 2 * block - 1) / (2 * block);   // 2 points / lane
  sq_main_kernel<<<grid, block, 0, stream>>>(params, pts, (float*)d_out, npts, nsq);
}